// VectorQuantizer_87617332838936
// MI455X (gfx1250) — compile-verified
//
#include <hip/hip_runtime.h>

typedef __attribute__((ext_vector_type(2))) float v2f;
typedef __attribute__((ext_vector_type(4))) float v4f;
typedef __attribute__((ext_vector_type(8))) float v8f;
typedef __attribute__((address_space(3))) float lds_float;

#define M_TOTAL 32768   // 8*4096 rows of z
#define N_CODES 8192
#define DIM     256
#define LDA     68      // LDS row stride (floats): 16B-aligned rows, 4-bank row spread

// async copy: global (16B) -> LDS, tracked by ASYNCcnt, no VGPR round-trip
__device__ __forceinline__ void async_copy_b128(lds_float* dst, const float* src) {
    asm volatile("global_load_async_to_lds_b128 %0, %1, off"
                 :: "v"(dst), "v"(src) : "memory");
}
__device__ __forceinline__ void wait_async0() {
#if __has_builtin(__builtin_amdgcn_s_wait_asynccnt)
    __builtin_amdgcn_s_wait_asynccnt(0);
#else
    asm volatile("s_wait_asynccnt 0x0" ::: "memory");
#endif
}

// ---------------------------------------------------------------------------
// Kernel 1: zero workspace accumulators (ws is poisoned by harness)
// ---------------------------------------------------------------------------
__global__ void vq_init_kernel(int* __restrict__ count, float* __restrict__ loss_sum) {
    int i = blockIdx.x * blockDim.x + threadIdx.x;
    if (i < N_CODES) count[i] = 0;
    if (i == 0) *loss_sum = 0.0f;
}

// ---------------------------------------------------------------------------
// Kernel 2: e2[row] = sum(emb[row,:]^2). One wave32 per row, 8 rows per block.
// ---------------------------------------------------------------------------
__global__ __launch_bounds__(256) void vq_rownorm_kernel(const float* __restrict__ x,
                                                         float* __restrict__ out,
                                                         int nrows) {
    int row  = blockIdx.x * 8 + (threadIdx.x >> 5);
    int lane = threadIdx.x & 31;
    if (row >= nrows) return;
    const float* p = x + (size_t)row * DIM + lane * 8;
    v4f a = *(const v4f*)(p);
    v4f b = *(const v4f*)(p + 4);
    float s = 0.0f;
#pragma unroll
    for (int i = 0; i < 4; ++i) s += a[i] * a[i] + b[i] * b[i];
#pragma unroll
    for (int off = 16; off > 0; off >>= 1) s += __shfl_xor(s, off, 32);
    if (lane == 0) out[row] = s;
}

// ---------------------------------------------------------------------------
// Kernel 3: fused GEMM (z . emb^T via V_WMMA_F32_16X16X4_F32) + argmin.
// Block = 256 threads = 8 waves as 2(M)x4(N). Each wave register-blocks a
// 2x2 grid of 16x16 WMMA tiles (32x32 per wave): 1 LDS fragment load/WMMA.
// Staging uses GLOBAL_LOAD_ASYNC_TO_LDS_B128 (ASYNCcnt) -- no VGPR staging,
// full MLP during tile fill. Block tile: 64 rows x 128 codes per chunk.
// argmin_n(z2 - 2 dot + e2) == argmin_n(e2 - 2 dot): z2 dropped.
// ---------------------------------------------------------------------------
__global__ __launch_bounds__(256) void vq_argmin_kernel(const float* __restrict__ z,
                                                        const float* __restrict__ emb,
                                                        const float* __restrict__ e2,
                                                        float* __restrict__ out_idx_f,
                                                        int* __restrict__ count) {
    __shared__ float smem[64 * LDA + 128 * LDA];   // z-tile | emb-chunk (~51 KB)
    float* zs = smem;
    float* es = smem + 64 * LDA;
    lds_float* zs3 = (lds_float*)zs;
    lds_float* es3 = (lds_float*)es;

    const int tid  = threadIdx.x;
    const int wave = tid >> 5;        // 0..7
    const int lane = tid & 31;
    const int wm   = wave >> 2;       // 0..1  (M 32-row sub-tile)
    const int wn   = wave & 3;        // 0..3  (N 32-code sub-tile)
    const int l16  = lane & 15;
    const int khi  = lane >> 4;       // K pair select per WMMA A/B layout

    const int mbase = blockIdx.x * 64;

    // per-thread staging coordinates (16 float4 per 64-col row)
    const int srow = tid >> 4;        // 0..15
    const int scol = (tid & 15) * 4;  // float index 0..60

    // running argmin state: [mi][r] -> row (wm*32 + mi*16 + r + 8*(lane>>4))
    float best[2][8];
    int   bidx[2][8];
#pragma unroll
    for (int mi = 0; mi < 2; ++mi)
#pragma unroll
        for (int r = 0; r < 8; ++r) { best[mi][r] = 3.4e38f; bidx[mi][r] = 0; }

    for (int nbase = 0; nbase < N_CODES; nbase += 128) {
        if (nbase + 128 < N_CODES)   // hint next emb chunk toward the caches
            __builtin_prefetch(emb + (size_t)(nbase + 128) * DIM, 0, 0);

        v8f acc00 = {0.f,0.f,0.f,0.f,0.f,0.f,0.f,0.f};
        v8f acc01 = acc00, acc10 = acc00, acc11 = acc00;

        for (int kc = 0; kc < 4; ++kc) {
            const int kbase = kc * 64;
            __syncthreads();   // staging buffer free (previous readers done)
            // stage z tile: 64 rows x 64 cols, 4 async b128 per thread
#pragma unroll
            for (int i = 0; i < 4; ++i) {
                int row = srow + i * 16;
                async_copy_b128(zs3 + row * LDA + scol,
                                z + (size_t)(mbase + row) * DIM + kbase + scol);
            }
            // stage emb chunk: 128 rows x 64 cols, 8 async b128 per thread
#pragma unroll
            for (int i = 0; i < 8; ++i) {
                int row = srow + i * 16;
                async_copy_b128(es3 + row * LDA + scol,
                                emb + (size_t)(nbase + row) * DIM + kbase + scol);
            }
            wait_async0();     // our async writes to LDS landed
            __syncthreads();   // everyone's writes visible
            // 16 K-steps of 4: 4 fragment loads -> 4 WMMAs (2x2 register block)
#pragma unroll
            for (int ks = 0; ks < 16; ++ks) {
                int k = ks * 4 + khi * 2;
                v2f a0 = *(const v2f*)(zs + (wm * 32 +      l16) * LDA + k);
                v2f a1 = *(const v2f*)(zs + (wm * 32 + 16 + l16) * LDA + k);
                v2f b0 = *(const v2f*)(es + (wn * 32 +      l16) * LDA + k);
                v2f b1 = *(const v2f*)(es + (wn * 32 + 16 + l16) * LDA + k);
                acc00 = __builtin_amdgcn_wmma_f32_16x16x4_f32(
                          false, a0, false, b0, (short)0, acc00, false, false);
                acc01 = __builtin_amdgcn_wmma_f32_16x16x4_f32(
                          false, a0, false, b1, (short)0, acc01, false, false);
                acc10 = __builtin_amdgcn_wmma_f32_16x16x4_f32(
                          false, a1, false, b0, (short)0, acc10, false, false);
                acc11 = __builtin_amdgcn_wmma_f32_16x16x4_f32(
                          false, a1, false, b1, (short)0, acc11, false, false);
            }
        }
        // epilogue: dist = e2[n] - 2*dot ; running argmin (n ascending => first-hit wins)
        const int   n0  = nbase + wn * 32 + l16;
        const int   n1  = n0 + 16;
        const float e20 = e2[n0];
        const float e21 = e2[n1];
#pragma unroll
        for (int r = 0; r < 8; ++r) {
            float d00 = e20 - 2.0f * acc00[r];
            float d01 = e21 - 2.0f * acc01[r];
            float d10 = e20 - 2.0f * acc10[r];
            float d11 = e21 - 2.0f * acc11[r];
            if (d00 < best[0][r]) { best[0][r] = d00; bidx[0][r] = n0; }
            if (d01 < best[0][r]) { best[0][r] = d01; bidx[0][r] = n1; }
            if (d10 < best[1][r]) { best[1][r] = d10; bidx[1][r] = n0; }
            if (d11 < best[1][r]) { best[1][r] = d11; bidx[1][r] = n1; }
        }
    }

    // reduce argmin across lanes/waves through LDS (staging buffer is free now)
    __syncthreads();
    float* redv = smem;                     // 4096 floats
    int*   redi = (int*)(smem + 4096);      // 4096 ints
#pragma unroll
    for (int mi = 0; mi < 2; ++mi)
#pragma unroll
        for (int r = 0; r < 8; ++r) {
            int slot = ((wave * 2 + mi) * 8 + r) * 32 + lane;
            redv[slot] = best[mi][r];
            redi[slot] = bidx[mi][r];
        }
    __syncthreads();
    if (tid < 64) {
        const int ml  = tid;                // local row 0..63
        const int mwm = ml >> 5;            // which M 32-row sub-tile
        const int mi  = (ml >> 4) & 1;      // which 16-row tile inside it
        const int r   = ml & 7;
        const int hi  = (ml >> 3) & 1;      // C layout: lanes16-31 hold M=r+8
        float bv = 3.4e38f;
        int   bn = 0x7fffffff;
        for (int w = 0; w < 4; ++w) {
            const int wv = mwm * 4 + w;
            for (int l = 0; l < 16; ++l) {
                const int slot = ((wv * 2 + mi) * 8 + r) * 32 + hi * 16 + l;
                float v  = redv[slot];
                int   nn = redi[slot];
                if (v < bv || (v == bv && nn < bn)) { bv = v; bn = nn; }
            }
        }
        out_idx_f[mbase + ml] = (float)bn;
        atomicAdd(&count[bn], 1);
    }
}

// ---------------------------------------------------------------------------
// Kernel 4: z_q = emb[idx]; write straight-through output; accumulate diff^2.
// One block per row (256 threads = 256 dims).
// ---------------------------------------------------------------------------
__global__ __launch_bounds__(256) void vq_gather_kernel(const float* __restrict__ z,
                                                        const float* __restrict__ emb,
                                                        const float* __restrict__ idxf,
                                                        float* __restrict__ zq_out,
                                                        float* __restrict__ loss_sum) {
    __shared__ float red[256];
    const int row = blockIdx.x;
    const int t   = threadIdx.x;
    const int n   = (int)idxf[row];
    float q  = emb[(size_t)n * DIM + t];
    float zz = z[(size_t)row * DIM + t];
    zq_out[(size_t)row * DIM + t] = q;      // z + sg(z_q - z) == z_q numerically
    float d = q - zz;
    red[t] = d * d;
    __syncthreads();
    for (int s = 128; s > 0; s >>= 1) {
        if (t < s) red[t] += red[t + s];
        __syncthreads();
    }
    if (t == 0) atomicAdd(loss_sum, red[0]);
}

// ---------------------------------------------------------------------------
// Kernel 5: age/usage update + losses.
// ---------------------------------------------------------------------------
__global__ void vq_finalize_kernel(const int* __restrict__ count,
                                   const float* __restrict__ age_in,
                                   const float* __restrict__ usage_in,
                                   const float* __restrict__ loss_sum,
                                   float* __restrict__ age_out,
                                   float* __restrict__ usage_out,
                                   float* __restrict__ losses) {
    int c = blockIdx.x * blockDim.x + threadIdx.x;
    if (c < N_CODES) {
        int k = count[c];
        age_out[c]   = (k > 0) ? 0.0f : age_in[c] + 1.0f;
        usage_out[c] = usage_in[c] + (float)k;
    }
    if (c == 0) {
        float mean = *loss_sum * (1.0f / ((float)M_TOTAL * (float)DIM));
        losses[0] = 0.25f * mean;   // loss_commit (BETA = 0.25)
        losses[1] = mean;           // loss_codebook
    }
}

// ---------------------------------------------------------------------------
extern "C" void kernel_launch(void* const* d_in, const int* in_sizes, int n_in,
                              void* d_out, int out_size, void* d_ws, size_t ws_size,
                              hipStream_t stream) {
    const float* z     = (const float*)d_in[0];   // [8,4096,256]
    const float* emb   = (const float*)d_in[1];   // [8192,256]
    const float* age   = (const float*)d_in[2];   // [8192]
    const float* usage = (const float*)d_in[3];   // [8192]

    float* out       = (float*)d_out;
    float* zq        = out;                         // 8388608
    float* losses    = out + 8388608;               // 2
    float* idxf      = out + 8388610;               // 32768
    float* age_out   = out + 8388610 + 32768;       // 8192
    float* usage_out = age_out + 8192;              // 8192

    char*  ws       = (char*)d_ws;
    int*   count    = (int*)ws;                     // 8192 ints
    float* e2       = (float*)(ws + 32768);         // 8192 floats
    float* loss_sum = (float*)(ws + 65536);         // 1 float

    vq_init_kernel<<<(N_CODES + 255) / 256, 256, 0, stream>>>(count, loss_sum);
    vq_rownorm_kernel<<<N_CODES / 8, 256, 0, stream>>>(emb, e2, N_CODES);
    vq_argmin_kernel<<<M_TOTAL / 64, 256, 0, stream>>>(z, emb, e2, idxf, count);
    vq_gather_kernel<<<M_TOTAL, 256, 0, stream>>>(z, emb, idxf, zq, loss_sum);
    vq_finalize_kernel<<<(N_CODES + 255) / 256, 256, 0, stream>>>(count, age, usage,
                                                                  loss_sum, age_out,
                                                                  usage_out, losses);
}